// VoxelProposalLayer_85968065397318
// MI455X (gfx1250) — compile-verified
//
#include <hip/hip_runtime.h>

// CDNA5 / gfx1250, wave32. Memory-bound voxel-occupancy scatter.
// No matrix structure -> no WMMA path applies; optimization is pure data
// movement: NT-hinted b128 streaming loads, L2-resident grid, predicated
// idempotent stores (no atomics).

typedef float v4f __attribute__((ext_vector_type(4)));
typedef int   v4i __attribute__((ext_vector_type(4)));

// Hardcoded constants from the reference
#define LO0 (0.0f)
#define LO1 (-25.6f)
#define LO2 (-2.0f)
#define HI0 (51.2f)
#define HI1 (25.6f)
#define HI2 (4.4f)
#define EPSM (1e-4f)

__device__ __forceinline__ void process_point(float x, float y, float z, int b,
                                              float* __restrict__ grid) {
    // mask: strict in-range with epsilon margin (matches reference exactly)
    bool m = (x > (LO0 + EPSM)) & (x < (HI0 - EPSM)) &
             (y > (LO1 + EPSM)) & (y < (HI1 - EPSM)) &
             (z > (LO2 + EPSM)) & (z < (HI2 - EPSM));
    // idx = trunc(ss * (p - lo) / (hi - lo)); IEEE division preserved
    float qx = (128.0f * (x - LO0)) / (HI0 - LO0);
    float qy = (128.0f * (y - LO1)) / (HI1 - LO1);
    float qz = (16.0f  * (z - LO2)) / (HI2 - LO2);
    int ix = (int)qx;
    int iy = (int)qy;
    int iz = (int)qz;
    ix = min(max(ix, 0), 127);
    iy = min(max(iy, 0), 127);
    iz = min(max(iz, 0), 15);
    if (m) {
        // All writers write 1.0f: idempotent, race-free, no atomic needed.
        int flat = (((b << 7) + ix) << 7 | iy) * 16 + iz;
        grid[flat] = 1.0f;
    }
}

__global__ void __launch_bounds__(256)
voxel_zero_kernel(float* __restrict__ grid, int n_vec, int total) {
    int t = blockIdx.x * blockDim.x + threadIdx.x;
    if (t < n_vec) {
        v4f z = {0.0f, 0.0f, 0.0f, 0.0f};
        ((v4f*)grid)[t] = z;
    }
    if (t == 0) {
        for (int i = 4 * n_vec; i < total; ++i) grid[i] = 0.0f;
    }
}

__global__ void __launch_bounds__(256)
voxel_scatter_kernel(const float* __restrict__ pts,
                     const int* __restrict__ bidx,
                     float* __restrict__ grid,
                     int n_quad, int n_pts) {
    int t = blockIdx.x * blockDim.x + threadIdx.x;
    if (t < n_quad) {
        // 4 points = 12 floats = three 16B-aligned b128 loads, NT-hinted so the
        // one-shot 128MB stream does not evict the L2-resident 8MB grid.
        const v4f* p4 = (const v4f*)pts;
        v4f a = __builtin_nontemporal_load(&p4[3 * t + 0]);
        v4f b = __builtin_nontemporal_load(&p4[3 * t + 1]);
        v4f c = __builtin_nontemporal_load(&p4[3 * t + 2]);
        v4i bi = __builtin_nontemporal_load(&((const v4i*)bidx)[t]);
        process_point(a.x, a.y, a.z, bi.x, grid);
        process_point(a.w, b.x, b.y, bi.y, grid);
        process_point(b.z, b.w, c.x, bi.z, grid);
        process_point(c.y, c.z, c.w, bi.w, grid);
    } else if (t == n_quad) {
        // scalar tail (N % 4 points), handled by one extra thread
        for (int i = 4 * n_quad; i < n_pts; ++i) {
            float x = pts[3 * i + 0];
            float y = pts[3 * i + 1];
            float z = pts[3 * i + 2];
            process_point(x, y, z, bidx[i], grid);
        }
    }
}

extern "C" void kernel_launch(void* const* d_in, const int* in_sizes, int n_in,
                              void* d_out, int out_size, void* d_ws, size_t ws_size,
                              hipStream_t stream) {
    const float* pts  = (const float*)d_in[0];  // [N,3] float32
    const int*   bidx = (const int*)d_in[1];    // [N] int32 (JAX x64-off)
    float*       out  = (float*)d_out;          // [B,128,128,16] float32

    int n_pts  = in_sizes[0] / 3;
    int n_quad = n_pts / 4;

    // 1) zero the occupancy grid (stream-ordered before scatter)
    int n_vec = out_size / 4;
    int zb = (n_vec + 255) / 256;
    if (zb < 1) zb = 1;
    voxel_zero_kernel<<<zb, 256, 0, stream>>>(out, n_vec, out_size);

    // 2) predicated scatter of 1.0f
    int threads_needed = n_quad + 1;  // +1 thread for the scalar tail
    int sb = (threads_needed + 255) / 256;
    voxel_scatter_kernel<<<sb, 256, 0, stream>>>(pts, bidx, out, n_quad, n_pts);
}